// TokenPruner_80762565034468
// MI455X (gfx1250) — compile-verified
//
#include <hip/hip_runtime.h>
#include <hip/hip_bf16.h>

typedef __attribute__((ext_vector_type(2))) float v2f;
typedef __attribute__((ext_vector_type(4))) float v4f;
typedef __attribute__((ext_vector_type(8))) float v8f;

#define B_DIM 32
#define N_DIM 4096
#define D_DIM 768
#define K_KEEP 2867   // round(4096 * 0.7)

// ---------------------------------------------------------------------------
// Kernel 1: scores[b][n] = dot(x[b][n][:], w[:]) via V_WMMA_F32_16X16X4_F32.
// One wave32 computes scores for a tile of 16 tokens, accumulating over
// D=768 in K=4 chunks (192 WMMA ops). Bandwidth-bound; WMMA rides for free.
//
// A (16x4 f32) per ISA: lanes 0-15 hold K={0,1}, lanes 16-31 hold K={2,3},
// one K per VGPR. B (4x16 f32, replicated w over N) is loaded with the same
// per-lane index math, making the result robust to K-slot permutation.
// ---------------------------------------------------------------------------
__global__ __launch_bounds__(256)
void tp_scores_wmma(const float* __restrict__ x, const float* __restrict__ w,
                    float* __restrict__ scores) {
  const int wave  = (blockIdx.x * blockDim.x + threadIdx.x) >> 5;  // tile id
  const int lane  = threadIdx.x & 31;
  const int tiles_per_batch = N_DIM / 16;                 // 256
  const int b  = wave / tiles_per_batch;
  const int t0 = (wave % tiles_per_batch) * 16;

  const int m    = lane & 15;          // token row within tile (A's M)
  const int koff = (lane >> 4) * 2;    // 0 for lanes 0-15, 2 for lanes 16-31

  const float* xrow = x + ((size_t)b * N_DIM + (t0 + m)) * D_DIM;

  v8f c = {};
#pragma unroll 8
  for (int k0 = 0; k0 < D_DIM; k0 += 4) {
    v2f a, bm;
    // A: token m, features k0+koff, k0+koff+1 (b64 load)
    a  = *(const v2f*)(xrow + k0 + koff);
    // B: w replicated across N columns — same per-slot k as A
    bm = *(const v2f*)(w + k0 + koff);
    c = __builtin_amdgcn_wmma_f32_16x16x4_f32(
        /*neg_a=*/false, a, /*neg_b=*/false, bm,
        /*c_mod=*/(short)0, c, /*reuse_a=*/false, /*reuse_b=*/false);
  }

  // C/D layout: lane 0 v[r] = score(token r), lane 16 v[r] = score(token 8+r)
  float* srow = scores + (size_t)b * N_DIM + t0;
  if (lane == 0) {
#pragma unroll
    for (int r = 0; r < 8; ++r) srow[r] = c[r];
  } else if (lane == 16) {
#pragma unroll
    for (int r = 0; r < 8; ++r) srow[8 + r] = c[r];
  }
}

// ---------------------------------------------------------------------------
// Kernel 2: per-batch bitonic sort of 4096 (score, index) pairs in LDS,
// descending by score, ties broken by ascending index (matches jax top_k).
// Emits the first K_KEEP indices. One 1024-thread block per batch; 32 KB LDS.
// ---------------------------------------------------------------------------
__device__ __forceinline__ bool tp_before(float va, int ia, float vb, int ib) {
  // true if (va,ia) should precede (vb,ib) in descending order
  return (va > vb) || (va == vb && ia < ib);
}

__global__ __launch_bounds__(1024)
void tp_sort_topk(const float* __restrict__ scores, int* __restrict__ idx_out) {
  __shared__ float sv[N_DIM];
  __shared__ int   si[N_DIM];
  const int b   = blockIdx.x;
  const int tid = threadIdx.x;

  const float* srow = scores + (size_t)b * N_DIM;
#pragma unroll
  for (int j = tid; j < N_DIM; j += 1024) {
    sv[j] = srow[j];
    si[j] = j;
  }
  __syncthreads();

  for (int size = 2; size <= N_DIM; size <<= 1) {
    for (int stride = size >> 1; stride > 0; stride >>= 1) {
      for (int t = tid; t < N_DIM / 2; t += 1024) {
        const int lo = 2 * t - (t & (stride - 1));
        const int hi = lo + stride;
        const bool desc = ((lo & size) == 0);
        float vl = sv[lo], vh = sv[hi];
        int   il = si[lo], ih = si[hi];
        const bool ordered = tp_before(vl, il, vh, ih);
        if (ordered != desc) {  // swap
          sv[lo] = vh; si[lo] = ih;
          sv[hi] = vl; si[hi] = il;
        }
      }
      __syncthreads();
    }
  }

  int* orow = idx_out + (size_t)b * K_KEEP;
  for (int j = tid; j < K_KEEP; j += 1024) orow[j] = si[j];
}

// ---------------------------------------------------------------------------
// Kernel 3: gather. out[b][j][:] = x[b][idx[b][j]][:].
// One 192-thread block per output row; b128 traffic via clang ext-vector
// float4 so nontemporal builtins accept it. Non-temporal stores: the 282 MB
// output is write-once, don't thrash the 192 MB L2.
// ---------------------------------------------------------------------------
__global__ __launch_bounds__(192)
void tp_gather(const float* __restrict__ x, const int* __restrict__ idx,
               float* __restrict__ out) {
  const int row = blockIdx.x;            // 0 .. B*K_KEEP-1
  const int b   = row / K_KEEP;
  const int j   = row - b * K_KEEP;
  const int tok = idx[(size_t)b * K_KEEP + j];   // uniform -> scalar load

  const v4f* src = (const v4f*)(x + ((size_t)b * N_DIM + tok) * D_DIM);
  v4f*       dst = (v4f*)(out + (size_t)row * D_DIM);

  const v4f v = src[threadIdx.x];               // 192 * 16B = 3 KB row
  __builtin_nontemporal_store(v, dst + threadIdx.x);
}

// ---------------------------------------------------------------------------
extern "C" void kernel_launch(void* const* d_in, const int* in_sizes, int n_in,
                              void* d_out, int out_size, void* d_ws, size_t ws_size,
                              hipStream_t stream) {
  const float* x = (const float*)d_in[0];   // [32, 4096, 768]
  const float* w = (const float*)d_in[1];   // [768, 1]
  // d_in[2] = bias: uniform shift, does not affect ranking or output -> unused
  float* out = (float*)d_out;               // [32, 2867, 768]

  float* ws_scores = (float*)d_ws;                          // 32*4096 floats
  int*   ws_idx    = (int*)((char*)d_ws + (size_t)B_DIM * N_DIM * sizeof(float));

  // 1) scores: 8192 tiles of 16 tokens, 8 waves per 256-thread block
  {
    const int tiles = B_DIM * (N_DIM / 16);   // 8192
    dim3 grid(tiles / 8), block(256);
    tp_scores_wmma<<<grid, block, 0, stream>>>(x, w, ws_scores);
  }
  // 2) per-batch bitonic top-k
  {
    dim3 grid(B_DIM), block(1024);
    tp_sort_topk<<<grid, block, 0, stream>>>(ws_scores, ws_idx);
  }
  // 3) gather selected rows
  {
    dim3 grid(B_DIM * K_KEEP), block(192);
    tp_gather<<<grid, block, 0, stream>>>(x, ws_idx, out);
  }
}